// AttentionNet_60876866454289
// MI455X (gfx1250) — compile-verified
//
#include <hip/hip_runtime.h>

// ---------------------------------------------------------------------------
// AttentionNet on MI455X (gfx1250, wave32, WMMA)
//   Stage 1: Xs = (X @ W1^T + b1) * w2 ;  Yf = Y @ W1^T + b1      (tiny GEMM)
//   Stage 2: out[bh,i,j] = relu( sum_d Xs[bh,i,d]*Yf[bh,j,d] + b2 )
//            = batched 1024x1024x64 GEMM via V_WMMA_F32_16X16X4_F32
// ---------------------------------------------------------------------------

#define BH   32
#define LSEQ 1024
#define DDIM 64

typedef __attribute__((ext_vector_type(2))) float v2f;
typedef __attribute__((ext_vector_type(8))) float v8f;

// ---------------- Stage 1: fused linear layer (negligible FLOPs) -----------
__global__ __launch_bounds__(256) void lin1_kernel(
    const float* __restrict__ X, const float* __restrict__ Y,
    const float* __restrict__ W1, const float* __restrict__ b1,
    const float* __restrict__ w2,
    float* __restrict__ Xs, float* __restrict__ Yf)
{
    __shared__ float sW[DDIM * 65];   // stride 65 -> conflict-free column reads
    __shared__ float sIn[4 * DDIM];

    const int  tid   = threadIdx.x;          // 0..255
    const bool isX   = (blockIdx.y == 0);
    const float* src = isX ? X : Y;
    float*       dst = isX ? Xs : Yf;
    const size_t rowbase = (size_t)blockIdx.x * 4;   // 4 rows per block

    // Stage W1 (64x64 f32) into LDS with padding
#pragma unroll
    for (int i = 0; i < 16; ++i) {
        int idx = tid + 256 * i;             // 0..4095
        int dd  = idx >> 6, ee = idx & 63;
        sW[dd * 65 + ee] = W1[idx];
    }
    // Stage 4 input rows
    sIn[tid] = src[rowbase * DDIM + tid];
    __syncthreads();

    const int d = tid & 63;                  // output feature
    const int r = tid >> 6;                  // row within block (0..3)
    float acc = b1[d];
#pragma unroll
    for (int e = 0; e < DDIM; ++e)
        acc = fmaf(sIn[r * DDIM + e], sW[d * 65 + e], acc);
    if (isX) acc *= w2[d];                   // fold w2 into the X side
    dst[(rowbase + r) * DDIM + d] = acc;
}

// ---------------- Stage 2: batched scores GEMM via f32 WMMA ----------------
// One wave per 64x64 output tile; 16 ksteps of 4, 16 WMMA per kstep.
__global__ __launch_bounds__(32) void scores_kernel(
    const float* __restrict__ Xs, const float* __restrict__ Yf,
    const float* __restrict__ b2p, float* __restrict__ out)
{
    const int lane = threadIdx.x;            // 0..31, full wave (EXEC all 1s)
    const int nt   = blockIdx.x;             // N tile (0..15)
    const int mt   = blockIdx.y;             // M tile (0..15)
    const int bh   = blockIdx.z;             // batch*head (0..31)

    const float* Abase = Xs + ((size_t)bh * LSEQ + (size_t)mt * 64) * DDIM;
    const float* Bbase = Yf + ((size_t)bh * LSEQ + (size_t)nt * 64) * DDIM;

    const int row = lane & 15;               // M (for A) / N (for B) within 16
    const int kh  = lane >> 4;               // half-wave selects K pair {0,1}/{2,3}

    v8f acc[4][4] = {};                      // 16 accumulators = 128 VGPRs

    for (int k0 = 0; k0 < DDIM; k0 += 4) {
        const int col = k0 + 2 * kh;         // per-lane K start (2 consecutive)
        v2f a[4], b[4];
#pragma unroll
        for (int i = 0; i < 4; ++i)          // A: 16x4 f32, lanes hold rows
            a[i] = *(const v2f*)(Abase + (size_t)(row + 16 * i) * DDIM + col);
#pragma unroll
        for (int j = 0; j < 4; ++j)          // B = Yf^T tile: lane n reads Yf row n
            b[j] = *(const v2f*)(Bbase + (size_t)(row + 16 * j) * DDIM + col);
#pragma unroll
        for (int i = 0; i < 4; ++i)
#pragma unroll
            for (int j = 0; j < 4; ++j)
                acc[i][j] = __builtin_amdgcn_wmma_f32_16x16x4_f32(
                    /*neg_a=*/false, a[i], /*neg_b=*/false, b[j],
                    /*c_mod=*/(short)0, acc[i][j],
                    /*reuse_a=*/false, /*reuse_b=*/false);
    }

    const float b2 = b2p[0];
    // C/D layout: VGPR r -> lanes 0-15: M=r, N=lane; lanes 16-31: M=r+8, N=lane-16
#pragma unroll
    for (int i = 0; i < 4; ++i) {
#pragma unroll
        for (int j = 0; j < 4; ++j) {
#pragma unroll
            for (int r = 0; r < 8; ++r) {
                const int m = mt * 64 + 16 * i + 8 * kh + r;
                const int n = nt * 64 + 16 * j + row;
                float v = acc[i][j][r] + b2;
                v = v > 0.0f ? v : 0.0f;
                out[((size_t)bh << 20) + (size_t)m * LSEQ + n] = v;
            }
        }
    }
}

// ---------------------------------------------------------------------------
extern "C" void kernel_launch(void* const* d_in, const int* in_sizes, int n_in,
                              void* d_out, int out_size, void* d_ws, size_t ws_size,
                              hipStream_t stream) {
    const float* X  = (const float*)d_in[0];
    const float* Y  = (const float*)d_in[1];
    const float* W1 = (const float*)d_in[2];
    const float* b1 = (const float*)d_in[3];
    const float* w2 = (const float*)d_in[4];
    const float* b2 = (const float*)d_in[5];
    float* out = (float*)d_out;

    float* Xs = (float*)d_ws;                           // 32*1024*64 f32 = 8 MB
    float* Yf = Xs + (size_t)BH * LSEQ * DDIM;          // next 8 MB

    // Stage 1: 32768 rows per input, 4 rows/block; grid.y selects X vs Y
    dim3 g1((BH * LSEQ) / 4, 2, 1);
    lin1_kernel<<<g1, 256, 0, stream>>>(X, Y, W1, b1, w2, Xs, Yf);

    // Stage 2: one wave per 64x64 tile of each 1024x1024 score matrix
    dim3 g2(LSEQ / 64, LSEQ / 64, BH);
    scores_kernel<<<g2, 32, 0, stream>>>(Xs, Yf, b2, out);
}